// PhotoWCT_80161269612985
// MI455X (gfx1250) — compile-verified
//
#include <hip/hip_runtime.h>
#include <hip/hip_bf16.h>

// ---------------------------------------------------------------------------
// PhotoWCT segmented whitening-coloring transform for MI455X (gfx1250).
// All GEMM work (segmented Grams, per-pixel 64x64 transform) runs on
// v_wmma_f32_16x16x4_f32 (CDNA5 fp32 WMMA) to match the fp32 reference.
// ---------------------------------------------------------------------------

typedef __attribute__((ext_vector_type(2))) float v2f;
typedef __attribute__((ext_vector_type(8))) float v8f;

#define NPIX (512 * 512)
#define NCH  64
#define NLAB 8
#define EPSV 1e-5f

// workspace layout (float offsets)
#define OFF_GRAM 0
#define GRAM_SZ  (2 * NLAB * 80 * 80)      // 102400: [feat][label][80][80]
#define OFF_COV  (OFF_GRAM + GRAM_SZ)      // 16 * 4096: [feat*8+l][64][64]
#define OFF_MU   (OFF_COV + 16 * 4096)     // 16 * 64
#define OFF_CNT  (OFF_MU + 16 * 64)        // 16
#define OFF_M    (OFF_CNT + 16)            // 16 * 4096: Wh (feat0) / Co (feat1)
#define OFF_T    (OFF_M + 16 * 4096)       // 8 * 4096
#define OFF_B    (OFF_T + 8 * 4096)        // 8 * 64
#define OFF_VAL  (OFF_B + 8 * 64)          // 8
#define WS_FLOATS (OFF_VAL + 8)            // ~267800 floats (~1.1 MB)

static __device__ __forceinline__ v8f wmma_f32_16x16x4(v2f a, v2f b, v8f c) {
  // 8-arg pattern: (neg_a, A, neg_b, B, c_mod, C, reuse_a, reuse_b)
  return __builtin_amdgcn_wmma_f32_16x16x4_f32(false, a, false, b, (short)0, c,
                                               false, false);
}

static __device__ __forceinline__ v8f zero8() {
  v8f z;
#pragma unroll
  for (int i = 0; i < 8; ++i) z[i] = 0.f;
  return z;
}

// ---------------------------------------------------------------------------
__global__ void zero_kernel(float* __restrict__ ws) {
  int i = blockIdx.x * blockDim.x + threadIdx.x;
  if (i < GRAM_SZ) ws[OFF_GRAM + i] = 0.f;
}

// ---------------------------------------------------------------------------
// Segmented augmented Gram: S[feat][l] = Xaug_l * Xaug_l^T over masked pixels.
// Xaug = [x(64ch); 1; 0...0] (80 channels = 5 WMMA row tiles). Row 64 of S
// gives per-channel masked sums, S[64][64] gives the masked pixel count.
// Wave w of each 256-thread block owns label w; 15 upper-triangle 16x16
// tiles are accumulated in VGPRs across all pixel chunks, flushed once.
#define GP 128
__global__ void __launch_bounds__(256) gram_kernel(
    const float* __restrict__ Xc, const float* __restrict__ Xs,
    const int* __restrict__ segc, const int* __restrict__ segs,
    float* __restrict__ ws) {
  const int feat = blockIdx.y;
  const float* X = feat ? Xs : Xc;
  const int* seg = feat ? segs : segc;
  float* gramf = ws + OFF_GRAM + feat * NLAB * 80 * 80;

  __shared__ float xs[NCH * (GP + 1)];  // padded rows -> conflict-light gathers
  __shared__ int sg[GP];

  const int tid = threadIdx.x;
  const int wave = tid >> 5;  // == label for this wave
  const int lane = tid & 31;
  const int m = lane & 15;
  const int kh = (lane >> 4) * 2;  // A/B K layout: lanes16-31 hold K=2,3

  const int TIa[15] = {0, 0, 0, 0, 0, 1, 1, 1, 1, 2, 2, 2, 3, 3, 4};
  const int TJa[15] = {0, 1, 2, 3, 4, 1, 2, 3, 4, 2, 3, 4, 3, 4, 4};

  v8f acc[15];
#pragma unroll
  for (int j = 0; j < 15; ++j) acc[j] = zero8();

  for (int ch = blockIdx.x; ch < NPIX / GP; ch += gridDim.x) {
    const int pbase = ch * GP;
    for (int idx = tid; idx < NCH * GP; idx += 256) {
      int c = idx >> 7, p = idx & (GP - 1);
      xs[c * (GP + 1) + p] = X[c * NPIX + pbase + p];
    }
    for (int idx = tid; idx < GP; idx += 256) sg[idx] = seg[pbase + idx];
    __syncthreads();

    for (int k = 0; k < GP / 4; ++k) {
      int p0 = k * 4 + kh;
      int p1 = p0 + 1;
      float m0 = (sg[p0] == wave) ? 1.f : 0.f;
      float m1 = (sg[p1] == wave) ? 1.f : 0.f;
      v2f f[5];
#pragma unroll
      for (int t = 0; t < 4; ++t) {
        int c = t * 16 + m;
        f[t].x = m0 * xs[c * (GP + 1) + p0];
        f[t].y = m1 * xs[c * (GP + 1) + p1];
      }
      // tile 4: channel 64 is the constant-1 (masked) row; 65..79 are zero
      f[4].x = (m == 0) ? m0 : 0.f;
      f[4].y = (m == 0) ? m1 : 0.f;
#pragma unroll
      for (int j = 0; j < 15; ++j)
        acc[j] = wmma_f32_16x16x4(f[TIa[j]], f[TJa[j]], acc[j]);
    }
    __syncthreads();
  }

  // flush partial tiles (C/D layout: VGPR r -> M=r (+8 for hi lanes), N=lane)
#pragma unroll
  for (int j = 0; j < 15; ++j) {
#pragma unroll
    for (int r = 0; r < 8; ++r) {
      int row = TIa[j] * 16 + r + (lane >> 4) * 8;
      int col = TJa[j] * 16 + m;
      atomicAdd(&gramf[wave * 6400 + row * 80 + col], acc[j][r]);
    }
  }
}

// ---------------------------------------------------------------------------
// cov[fl] = (S - n mu mu^T)/div (+ I for content), mu/cnt extracted from the
// augmented Gram (row 64 / corner). One block per (feat,label).
__global__ void __launch_bounds__(256) cov_kernel(float* __restrict__ ws) {
  const int fl = blockIdx.x;
  const int feat = fl >> 3;
  const float* S = ws + OFF_GRAM + fl * 6400;
  __shared__ float muL[64];
  __shared__ float nsh;
  const int tid = threadIdx.x;
  if (tid == 0) {
    nsh = S[64 * 80 + 64];
    ws[OFF_CNT + fl] = nsh;
  }
  __syncthreads();
  if (tid < 64) {
    float mu = S[tid * 80 + 64] / fmaxf(nsh, 1.f);
    muL[tid] = mu;
    ws[OFF_MU + fl * 64 + tid] = mu;
  }
  __syncthreads();
  const float n = nsh;
  const float div = (n - 1.f == 0.f) ? EPSV : (n - 1.f);
  for (int idx = tid; idx < 4096; idx += 256) {
    int a = idx >> 6, b = idx & 63;
    // only upper-triangle tiles were accumulated; use symmetry
    float v = ((a >> 4) <= (b >> 4)) ? S[a * 80 + b] : S[b * 80 + a];
    float cov = (v - n * muL[a] * muL[b]) / div;
    if (feat == 0 && a == b) cov += 1.f;
    ws[OFF_COV + fl * 4096 + idx] = cov;
  }
}

// ---------------------------------------------------------------------------
// Parallel-order cyclic Jacobi eigensolver (64x64 symmetric) in LDS.
// 32 disjoint rotations per round (round-robin schedule), 63 rounds/sweep.
// Emits M = V f(Lambda) V^T with the reference eigenvalue clamp.
static __device__ __forceinline__ int jpos(int i, int r) {
  return (i == 0) ? 0 : (((i - 1 + r) % 63) + 1);
}

__global__ void __launch_bounds__(256) jacobi_kernel(float* __restrict__ ws) {
  const int fl = blockIdx.x;
  const int feat = fl >> 3;
  const float* covp = ws + OFF_COV + fl * 4096;
  float* Mout = ws + OFF_M + fl * 4096;
  __shared__ float A[64 * 65];
  __shared__ float V[64 * 65];
  __shared__ float cc[32], ss[32];
  __shared__ int pp[32], qq[32];
  __shared__ float dspec[64];
  const int tid = threadIdx.x;

  for (int idx = tid; idx < 4096; idx += 256) {
    int a = idx >> 6, b = idx & 63;
    A[a * 65 + b] = covp[idx];
    V[a * 65 + b] = (a == b) ? 1.f : 0.f;
  }
  __syncthreads();

  for (int sweep = 0; sweep < 10; ++sweep) {
    for (int r = 0; r < 63; ++r) {
      if (tid < 32) {
        int p = jpos(tid, r), q = jpos(63 - tid, r);
        if (p > q) { int t = p; p = q; q = t; }
        float app = A[p * 65 + p], aqq = A[q * 65 + q], apq = A[p * 65 + q];
        float c = 1.f, s = 0.f;
        if (fabsf(apq) > 1e-20f) {
          float tau = (aqq - app) / (2.f * apq);
          float t = copysignf(1.f, tau) / (fabsf(tau) + sqrtf(1.f + tau * tau));
          c = 1.f / sqrtf(1.f + t * t);
          s = t * c;
        }
        cc[tid] = c; ss[tid] = s; pp[tid] = p; qq[tid] = q;
      }
      __syncthreads();
      // stage 1: rotate rows p,q (disjoint across the 32 pairs)
      for (int idx = tid; idx < 2048; idx += 256) {
        int k = idx >> 6, j = idx & 63;
        int p = pp[k], q = qq[k];
        float c = cc[k], s = ss[k];
        float x = A[p * 65 + j], y = A[q * 65 + j];
        A[p * 65 + j] = c * x - s * y;
        A[q * 65 + j] = s * x + c * y;
      }
      __syncthreads();
      // stage 2: rotate cols p,q (+ eigenvector columns), disjoint again
      for (int idx = tid; idx < 2048; idx += 256) {
        int k = idx >> 6, j = idx & 63;
        int p = pp[k], q = qq[k];
        float c = cc[k], s = ss[k];
        float x = A[j * 65 + p], y = A[j * 65 + q];
        A[j * 65 + p] = c * x - s * y;
        A[j * 65 + q] = s * x + c * y;
        float vx = V[j * 65 + p], vy = V[j * 65 + q];
        V[j * 65 + p] = c * vx - s * vy;
        V[j * 65 + q] = s * vx + c * vy;
      }
      __syncthreads();
    }
  }

  if (tid < 64) {
    float e = A[tid * 65 + tid];
    float d = 0.f;
    if (e >= EPSV) d = (feat == 0) ? rsqrtf(e) : sqrtf(e);
    dspec[tid] = d;
  }
  __syncthreads();
  for (int idx = tid; idx < 4096; idx += 256) {
    int a = idx >> 6, b = idx & 63;
    float sum = 0.f;
    for (int i = 0; i < 64; ++i) sum += V[a * 65 + i] * dspec[i] * V[b * 65 + i];
    Mout[idx] = sum;
  }
}

// ---------------------------------------------------------------------------
// T_l = Co_l * Wh_l ; bias_l = mu_s - T_l mu_c ; validity flags.
__global__ void __launch_bounds__(256) tmat_kernel(float* __restrict__ ws) {
  const int l = blockIdx.x;
  const float* Wh = ws + OFF_M + l * 4096;
  const float* Co = ws + OFF_M + (8 + l) * 4096;
  __shared__ float Tl[64 * 65];
  __shared__ float muc[64], mus[64];
  const int tid = threadIdx.x;
  if (tid < 64) {
    muc[tid] = ws[OFF_MU + l * 64 + tid];
    mus[tid] = ws[OFF_MU + (8 + l) * 64 + tid];
  }
  for (int idx = tid; idx < 4096; idx += 256) {
    int a = idx >> 6, b = idx & 63;
    float s = 0.f;
    for (int i = 0; i < 64; ++i) s += Co[a * 64 + i] * Wh[i * 64 + b];
    Tl[a * 65 + b] = s;
  }
  __syncthreads();
  for (int idx = tid; idx < 4096; idx += 256)
    ws[OFF_T + l * 4096 + idx] = Tl[(idx >> 6) * 65 + (idx & 63)];
  if (tid < 64) {
    float s = 0.f;
    for (int i = 0; i < 64; ++i) s += Tl[tid * 65 + i] * muc[i];
    ws[OFF_B + l * 64 + tid] = mus[tid] - s;
  }
  if (tid == 0) {
    float nc = ws[OFF_CNT + l], ns = ws[OFF_CNT + 8 + l];
    bool v = (nc > 10.f) && (ns > 10.f) && (nc < 100.f * ns) &&
             (ns < 100.f * nc);
    ws[OFF_VAL + l] = v ? 1.f : 0.f;
  }
}

// ---------------------------------------------------------------------------
// Apply: out[:,p] = valid[l_p] ? T_{l_p}(x_p - mu_c) + mu_s : x_p.
// Implemented as Y = sum_l T_l (X .* mask_l) via WMMA (masked columns are
// zero, so one accumulator suffices), then per-pixel bias/passthrough.
#define AP 128
__global__ void __launch_bounds__(256) apply_kernel(
    const float* __restrict__ Xc, const int* __restrict__ seg,
    const float* __restrict__ ws, float* __restrict__ out) {
  __shared__ float xs[NCH * (AP + 1)];
  __shared__ float bl[NLAB * 64];
  __shared__ float TlL[16 * 68];
  __shared__ int sg[AP];
  __shared__ int pres[NLAB];
  __shared__ float vldL[NLAB];

  const int tid = threadIdx.x;
  const int pbase = blockIdx.x * AP;
  if (tid < NLAB) {
    pres[tid] = 0;
    vldL[tid] = ws[OFF_VAL + tid];
  }
  for (int idx = tid; idx < NCH * AP; idx += 256) {
    int c = idx >> 7, p = idx & (AP - 1);
    xs[c * (AP + 1) + p] = Xc[c * NPIX + pbase + p];
  }
  for (int idx = tid; idx < NLAB * 64; idx += 256) bl[idx] = ws[OFF_B + idx];
  if (tid < AP) sg[tid] = seg[pbase + tid];
  __syncthreads();
  if (tid < AP) pres[sg[tid]] = 1;  // benign same-value race
  __syncthreads();

  const int w = tid >> 5, lane = tid & 31;
  const int m = lane & 15;
  const int kh = (lane >> 4) * 2;
  const int pcol = w * 16;      // this wave's 16-pixel column tile
  const int myl = sg[pcol + m]; // label of this lane's pixel

  for (int rt = 0; rt < 4; ++rt) {
    v8f acc = zero8();
    for (int l = 0; l < NLAB; ++l) {
      bool use = (pres[l] != 0) && (vldL[l] != 0.f);  // block-uniform
      if (use) {
        for (int idx = tid; idx < 1024; idx += 256) {
          int r = idx >> 6, c = idx & 63;
          TlL[r * 68 + c] = ws[OFF_T + l * 4096 + (rt * 16 + r) * 64 + c];
        }
      }
      __syncthreads();
      if (use) {
        float msk = (myl == l) ? 1.f : 0.f;
#pragma unroll
        for (int k = 0; k < 16; ++k) {
          int c0 = k * 4 + kh;
          v2f a, b;
          a.x = TlL[m * 68 + c0];
          a.y = TlL[m * 68 + c0 + 1];
          b.x = msk * xs[c0 * (AP + 1) + pcol + m];
          b.y = msk * xs[(c0 + 1) * (AP + 1) + pcol + m];
          acc = wmma_f32_16x16x4(a, b, acc);
        }
      }
      __syncthreads();
    }
#pragma unroll
    for (int r = 0; r < 8; ++r) {
      int row = rt * 16 + r + (lane >> 4) * 8;
      float res = (vldL[myl] != 0.f) ? (acc[r] + bl[myl * 64 + row])
                                     : xs[row * (AP + 1) + pcol + m];
      out[row * NPIX + pbase + pcol + m] = res;
    }
  }
}

// ---------------------------------------------------------------------------
extern "C" void kernel_launch(void* const* d_in, const int* in_sizes, int n_in,
                              void* d_out, int out_size, void* d_ws,
                              size_t ws_size, hipStream_t stream) {
  const float* cont = (const float*)d_in[0];
  const float* styl = (const float*)d_in[1];
  const int* cseg = (const int*)d_in[2];
  const int* sseg = (const int*)d_in[3];
  // d_in[4] = n_labels, fixed at 8 per the reference setup.
  float* ws = (float*)d_ws;
  float* out = (float*)d_out;
  (void)in_sizes; (void)n_in; (void)out_size; (void)ws_size;

  zero_kernel<<<(GRAM_SZ + 255) / 256, 256, 0, stream>>>(ws);
  gram_kernel<<<dim3(256, 2), 256, 0, stream>>>(cont, styl, cseg, sseg, ws);
  cov_kernel<<<16, 256, 0, stream>>>(ws);
  jacobi_kernel<<<16, 256, 0, stream>>>(ws);
  tmat_kernel<<<8, 256, 0, stream>>>(ws);
  apply_kernel<<<NPIX / AP, 256, 0, stream>>>(cont, cseg, ws, out);
}